// Featurizer_61314953117884
// MI455X (gfx1250) — compile-verified
//
#include <hip/hip_runtime.h>
#include <math.h>

#define N_NODES 100000
#define N_EDGES 1000000
#define HIDDEN  128
#define GEPS    1e-6f
#define LN_EPS  1e-5f

typedef __attribute__((ext_vector_type(2))) float v2f;
typedef __attribute__((ext_vector_type(4))) float v4f;
typedef __attribute__((ext_vector_type(8))) float v8f;

#define FPAD 28   // padded feature row length in LDS (16B-aligned, reduced bank conflicts)

struct F3 { float x, y, z; };
__device__ __forceinline__ F3 f3sub(F3 a, F3 b) { return F3{a.x-b.x, a.y-b.y, a.z-b.z}; }
__device__ __forceinline__ float f3dot(F3 a, F3 b) { return a.x*b.x + a.y*b.y + a.z*b.z; }
__device__ __forceinline__ F3 f3cross(F3 a, F3 b) {
    return F3{a.y*b.z - a.z*b.y, a.z*b.x - a.x*b.z, a.x*b.y - a.y*b.x};
}
__device__ __forceinline__ F3 f3norm(F3 v) {
    float n = sqrtf(f3dot(v, v));
    float d = 1.0f / fmaxf(n, 1e-12f);
    return F3{v.x*d, v.y*d, v.z*d};
}
__device__ __forceinline__ float signf(float x) { return (x > 0.f) ? 1.f : ((x < 0.f) ? -1.f : 0.f); }

// xor-butterfly sum over 16-lane rows using v_permlane16_b32 (VALU-only, no DS pipe).
// Lane-select nibbles encode j^mask for j=0..15; same selects apply to both rows.
__device__ __forceinline__ float xsum16(float v) {
    unsigned u, p;
    u = __float_as_uint(v);
    p = __builtin_amdgcn_permlane16(u, u, 0x67452301u, 0xEFCDAB89u, false, false); // ^1
    v += __uint_as_float(p);
    u = __float_as_uint(v);
    p = __builtin_amdgcn_permlane16(u, u, 0x54761032u, 0xDCFE98BAu, false, false); // ^2
    v += __uint_as_float(p);
    u = __float_as_uint(v);
    p = __builtin_amdgcn_permlane16(u, u, 0x32107654u, 0xBA98FEDCu, false, false); // ^4
    v += __uint_as_float(p);
    u = __float_as_uint(v);
    p = __builtin_amdgcn_permlane16(u, u, 0xFEDCBA98u, 0x76543210u, false, false); // ^8
    v += __uint_as_float(p);
    return v;
}

// ---------------- Kernel A: per-node geometric features + frames ----------------
__global__ void feat_kernel(const float* __restrict__ X, const int* __restrict__ batch,
                            float* __restrict__ Qbuf, float* __restrict__ nfeat,
                            int* __restrict__ validbuf)
{
    int i = blockIdx.x * blockDim.x + threadIdx.x;
    if (i >= N_NODES) return;
    int im1 = max(i - 1, 0);
    int ip1 = min(i + 1, N_NODES - 1);
    int ip2 = min(i + 2, N_NODES - 1);

    int b_im1 = batch[im1], b_i = batch[i], b_ip1 = batch[ip1], b_ip2 = batch[ip2];
    bool bnd0 = (i >= 1) && (b_i != b_im1);
    bool bnd1 = (i + 1 < N_NODES) && (b_ip1 != b_i);
    bool bnd2 = (i + 2 < N_NODES) && (b_ip2 != b_ip1);
    bool bad_a = bnd0 || bnd1 || (i == 0) || (i == N_NODES - 1);
    bool bad_d = bnd0 || bnd1 || bnd2 || (i == 0) || (i == N_NODES - 2) || (i == N_NODES - 1);

    F3 Xm1 = {X[im1*3+0], X[im1*3+1], X[im1*3+2]};
    F3 X0  = {X[i*3+0],   X[i*3+1],   X[i*3+2]};
    F3 X1  = {X[ip1*3+0], X[ip1*3+1], X[ip1*3+2]};
    F3 X2  = {X[ip2*3+0], X[ip2*3+1], X[ip2*3+2]};

    F3 Up = f3norm(f3sub(X0, Xm1));   // U[i-1]
    F3 U0 = f3norm(f3sub(X1, X0));    // U[i]
    F3 U1 = f3norm(f3sub(X2, X1));    // U[i+1]

    // dihedral
    F3 c1 = f3norm(f3cross(Up, U0));
    F3 c2 = f3norm(f3cross(U0, U1));
    float cosd = fminf(fmaxf(f3dot(c1, c2), -1.0f + GEPS), 1.0f - GEPS);
    float dih = acosf(cosd) * signf(f3dot(c2, Up));
    float dsin = bad_d ? 0.f : sinf(dih);
    float dcos = bad_d ? 0.f : cosf(dih);

    // angle
    F3 d0 = f3norm(f3sub(Xm1, X0));
    F3 d1 = f3norm(f3sub(X1, X0));
    float cosa = f3dot(d0, d1);
    float sina = sqrtf(1.0f - cosa*cosa + GEPS);
    if (bad_a) { sina = 0.f; cosa = 0.f; }

    nfeat[(size_t)i*4 + 0] = dsin;
    nfeat[(size_t)i*4 + 1] = dcos;
    nfeat[(size_t)i*4 + 2] = sina;
    nfeat[(size_t)i*4 + 3] = cosa;

    // frame Q: columns [bvec, nvec, bvec x nvec]
    F3 bv = f3norm(f3sub(Up, U0));
    F3 nv = f3norm(f3cross(Up, U0));
    F3 cv = f3cross(bv, nv);
    float Q[9];
    Q[0]=bv.x; Q[1]=nv.x; Q[2]=cv.x;
    Q[3]=bv.y; Q[4]=nv.y; Q[5]=cv.y;
    Q[6]=bv.z; Q[7]=nv.z; Q[8]=cv.z;
#pragma unroll
    for (int j = 0; j < 9; j++) Qbuf[(size_t)i*9 + j] = bad_a ? 0.f : Q[j];
    validbuf[i] = bad_a ? 0 : 1;
}

// ---------------- Pack W into per-lane WMMA B-fragment layout ----------------
// Wpack[((c*NSLICE + s)*32 + lane)*2 + {0,1}] = B fragment (v2f) for chunk c, slice s.
// Fragment layout: lanes 0-15 hold rows {4s,4s+1}, lanes 16-31 rows {4s+2,4s+3}, N = lane&15.
__global__ void pack_kernel(const float* __restrict__ W, int K, int nslice,
                            float* __restrict__ Wpack)
{
    int idx = blockIdx.x * blockDim.x + threadIdx.x;
    int total = 8 * nslice * 32;
    if (idx >= total) return;
    int lane = idx & 31;
    int s    = (idx >> 5) % nslice;
    int c    = idx / (32 * nslice);
    int half = lane >> 4;
    int lcol = lane & 15;
    int col  = c * 16 + lcol;
    int k0   = 4 * s + 2 * half;
    Wpack[(size_t)idx * 2 + 0] = (k0     < K) ? W[(size_t)k0 * HIDDEN + col]       : 0.f;
    Wpack[(size_t)idx * 2 + 1] = (k0 + 1 < K) ? W[(size_t)(k0 + 1) * HIDDEN + col] : 0.f;
}

// ---------------- Shared: 32-row (two 16-row tiles) GEMM + LayerNorm + store ----------------
// A fragments already in WMMA layout; each B fragment feeds both tiles immediately.
template<int NSLICE>
__device__ __forceinline__ void gemm32_ln_store(
    const v2f (&a0)[NSLICE], const v2f (&a1)[NSLICE],
    const float* __restrict__ Wpack,
    const float* __restrict__ bias,
    const float* __restrict__ gamma,
    const float* __restrict__ beta,
    float* __restrict__ out,                    // row-major [rows][128]
    int row_base)                               // first of 32 rows handled by this wave
{
    const int lane = threadIdx.x & 31;
    const int half = lane >> 4;
    const int lcol = lane & 15;

    const v2f* __restrict__ bp = (const v2f*)Wpack;

    v8f acc0[8] = {};
    v8f acc1[8] = {};
#pragma unroll
    for (int c = 0; c < 8; c++) {
#pragma unroll
        for (int s = 0; s < NSLICE; s++) {
            v2f b = bp[(c * NSLICE + s) * 32 + lane];
            acc0[c] = __builtin_amdgcn_wmma_f32_16x16x4_f32(
                false, a0[s], false, b, (short)0, acc0[c], false, false);
            acc1[c] = __builtin_amdgcn_wmma_f32_16x16x4_f32(
                false, a1[s], false, b, (short)0, acc1[c], false, false);
        }
    }

    // bias (per column, same for both tiles)
#pragma unroll
    for (int c = 0; c < 8; c++) {
        float bc = bias[c*16 + lcol];
#pragma unroll
        for (int r = 0; r < 8; r++) { acc0[c][r] += bc; acc1[c][r] += bc; }
    }

    // LayerNorm + store, one tile at a time
#pragma unroll
    for (int t = 0; t < 2; t++) {
        v8f* acc = t ? acc1 : acc0;
        float mean[8], inv[8];
#pragma unroll
        for (int r = 0; r < 8; r++) {
            float s1 = 0.f, s2 = 0.f;
#pragma unroll
            for (int c = 0; c < 8; c++) { float v = acc[c][r]; s1 += v; s2 += v*v; }
            s1 = xsum16(s1);
            s2 = xsum16(s2);
            float mu  = s1 * (1.0f / HIDDEN);
            float var = s2 * (1.0f / HIDDEN) - mu*mu;
            mean[r] = mu;
            inv[r]  = 1.0f / sqrtf(var + LN_EPS);
        }
#pragma unroll
        for (int c = 0; c < 8; c++) {
            int col = c * 16 + lcol;
            float gc = gamma[col];
            float bt = beta[col];
#pragma unroll
            for (int r = 0; r < 8; r++) {
                int row = row_base + t * 16 + r + half * 8;
                float val = (acc[c][r] - mean[r]) * inv[r] * gc + bt;
                // streaming output: non-temporal, keep L2 for gather working set
                __builtin_nontemporal_store(val, &out[(size_t)row * HIDDEN + col]);
            }
        }
    }
}

// ---------------- Kernel B: node GEMM (K=4) + LN ----------------
__global__ void __launch_bounds__(256, 1)
node_kernel(const float* __restrict__ nfeat,
            const float* __restrict__ Wpack, const float* __restrict__ bn,
            const float* __restrict__ gn, const float* __restrict__ betan,
            float* __restrict__ out)
{
    int wave = (blockIdx.x * blockDim.x + threadIdx.x) >> 5;
    int lane = threadIdx.x & 31;
    int base = wave * 32;
    if (base >= N_NODES) return;   // wave-uniform -> EXEC stays all-1 for active waves

    int half = lane >> 4;
    int lcol = lane & 15;
    // A fragments straight from memory: row-major [node][4] is already fragment-friendly.
    v2f a0[1], a1[1];
    a0[0] = *(const v2f*)(nfeat + (size_t)(base + lcol)      * 4 + 2 * half);
    a1[0] = *(const v2f*)(nfeat + (size_t)(base + 16 + lcol) * 4 + 2 * half);

    gemm32_ln_store<1>(a0, a1, Wpack, bn, gn, betan, out, base);
}

// ---------------- Kernel C: edge features + GEMM (K=23 padded to 24) + LN ----------------
__global__ void __launch_bounds__(256, 1)
edge_kernel(const float* __restrict__ X, const int* __restrict__ eidx,
            const float* __restrict__ Qbuf, const int* __restrict__ validbuf,
            const float* __restrict__ Wpack, const float* __restrict__ be,
            const float* __restrict__ ge, const float* __restrict__ betae,
            float* __restrict__ out)
{
    __shared__ float smem[8 * 32 * FPAD];   // 28 KB: per-wave 32 feature rows (padded)

    int wave = (blockIdx.x * blockDim.x + threadIdx.x) >> 5;
    int lane = threadIdx.x & 31;
    int base = wave * 32;
    if (base >= N_EDGES) return;   // wave-uniform

    int e = base + lane;
    int s = eidx[e];
    int t = eidx[N_EDGES + e];

    float Qs[9], Qt[9];
#pragma unroll
    for (int j = 0; j < 9; j++) { Qs[j] = Qbuf[(size_t)s*9 + j]; Qt[j] = Qbuf[(size_t)t*9 + j]; }
    bool ok_s = validbuf[s] != 0;
    bool ok_t = validbuf[t] != 0;
    bool ok   = ok_s && ok_t;

    F3 Xs = {X[(size_t)s*3+0], X[(size_t)s*3+1], X[(size_t)s*3+2]};
    F3 Xt = {X[(size_t)t*3+0], X[(size_t)t*3+1], X[(size_t)t*3+2]};

    // R = Qt^T * Qs : R[i][k] = sum_j Qt[3j+i] * Qs[3j+k]
    float R[3][3];
#pragma unroll
    for (int i2 = 0; i2 < 3; i2++)
#pragma unroll
        for (int k2 = 0; k2 < 3; k2++)
            R[i2][k2] = Qt[i2]*Qs[k2] + Qt[3+i2]*Qs[3+k2] + Qt[6+i2]*Qs[6+k2];

    float Rxx = R[0][0], Ryy = R[1][1], Rzz = R[2][2];
    float q0 = signf(R[2][1] - R[1][2]) * 0.5f * sqrtf(fabsf(1.f + Rxx - Ryy - Rzz));
    float q1 = signf(R[0][2] - R[2][0]) * 0.5f * sqrtf(fabsf(1.f - Rxx + Ryy - Rzz));
    float q2 = signf(R[1][0] - R[0][1]) * 0.5f * sqrtf(fabsf(1.f - Rxx - Ryy + Rzz));
    float qw = 0.5f * sqrtf(fmaxf(1.f + Rxx + Ryy + Rzz, 0.f));
    float qn = sqrtf(q0*q0 + q1*q1 + q2*q2 + qw*qw);
    float qd = 1.0f / fmaxf(qn, 1e-12f);

    float f[24];
    f[0] = ok ? q0*qd : 0.f;
    f[1] = ok ? q1*qd : 0.f;
    f[2] = ok ? q2*qd : 0.f;
    f[3] = ok ? qw*qd : 0.f;

    F3 dv = f3sub(Xs, Xt);
    float dist = sqrtf(f3dot(dv, dv) + GEPS);
#pragma unroll
    for (int k = 0; k < 16; k++) {
        float mu = (20.0f / 15.0f) * (float)k;
        float d  = (dist - mu) * (16.0f / 20.0f);
        f[4 + k] = expf(-d*d);
    }

    F3 dn = f3norm(dv);
    float dir0 = Qt[0]*dn.x + Qt[3]*dn.y + Qt[6]*dn.z;
    float dir1 = Qt[1]*dn.x + Qt[4]*dn.y + Qt[7]*dn.z;
    float dir2 = Qt[2]*dn.x + Qt[5]*dn.y + Qt[8]*dn.z;
    f[20] = ok_t ? dir0 : 0.f;
    f[21] = ok_t ? dir1 : 0.f;
    f[22] = ok_t ? dir2 : 0.f;
    f[23] = 0.f;  // K padding to 24

    // Stage feature rows through LDS, read back in WMMA A-fragment layout.
    float* myrow = smem + (size_t)threadIdx.x * FPAD;
#pragma unroll
    for (int j = 0; j < 24; j += 4) {
        v4f tmp = {f[j], f[j+1], f[j+2], f[j+3]};
        *(v4f*)(myrow + j) = tmp;
    }
    __builtin_amdgcn_wave_barrier();   // same-wave DS ops are in-order; pin compiler ordering

    int half = lane >> 4;
    int lcol = lane & 15;
    const float* wbase = smem + (size_t)(threadIdx.x & ~31) * FPAD;
    v2f a0[6], a1[6];
#pragma unroll
    for (int sl = 0; sl < 6; sl++) {
        a0[sl] = *(const v2f*)(wbase + (size_t)lcol        * FPAD + 4*sl + 2*half);
        a1[sl] = *(const v2f*)(wbase + (size_t)(16 + lcol) * FPAD + 4*sl + 2*half);
    }

    gemm32_ln_store<6>(a0, a1, Wpack, be, ge, betae, out, base);
}

extern "C" void kernel_launch(void* const* d_in, const int* in_sizes, int n_in,
                              void* d_out, int out_size, void* d_ws, size_t ws_size,
                              hipStream_t stream)
{
    const float* X      = (const float*)d_in[0];
    const int*   batch  = (const int*)  d_in[1];
    const int*   eidx   = (const int*)  d_in[2];
    const float* Wn     = (const float*)d_in[3];
    const float* bn     = (const float*)d_in[4];
    const float* gn     = (const float*)d_in[5];
    const float* betan  = (const float*)d_in[6];
    const float* We     = (const float*)d_in[7];
    const float* be     = (const float*)d_in[8];
    const float* ge     = (const float*)d_in[9];
    const float* betae  = (const float*)d_in[10];
    float* out = (float*)d_out;

    float* Qbuf   = (float*)d_ws;                        // 100000*9 floats
    float* nfeat  = Qbuf + (size_t)N_NODES * 9;          // 100000*4 floats
    int*   valid  = (int*)(nfeat + (size_t)N_NODES*4);   // 100000 ints
    float* WpackN = (float*)(valid + N_NODES);           // 8*1*32*2 = 512 floats
    float* WpackE = WpackN + 512;                        // 8*6*32*2 = 3072 floats

    feat_kernel<<<(N_NODES + 255) / 256, 256, 0, stream>>>(X, batch, Qbuf, nfeat, valid);
    pack_kernel<<<1, 256, 0, stream>>>(Wn, 4, 1, WpackN);          // 256 threads = 8*1*32
    pack_kernel<<<6, 256, 0, stream>>>(We, 23, 6, WpackE);         // 1536 threads = 8*6*32

    // 100000 nodes / 32 per wave = 3125 waves; 8 waves/block -> 391 blocks
    node_kernel<<<391, 256, 0, stream>>>(nfeat, WpackN, bn, gn, betan, out);

    // 1000000 edges / 32 per wave = 31250 waves; 8 waves/block -> 3907 blocks
    edge_kernel<<<3907, 256, 0, stream>>>(X, eidx, Qbuf, valid,
                                          WpackE, be, ge, betae,
                                          out + (size_t)N_NODES * HIDDEN);
}